// ExampleConv2_22711787061742
// MI455X (gfx1250) — compile-verified
//
#include <hip/hip_runtime.h>
#include <stdint.h>

typedef __attribute__((ext_vector_type(2))) float v2f;
typedef __attribute__((ext_vector_type(8))) float v8f;

#define D 128            // D_IN == D_OUT == 128
#define ROWS_PER_BLOCK 128
#define THREADS 256

// -------------------------------------------------------------------------
// Kernel 1: h = x @ W  via V_WMMA_F32_16X16X4_F32.
// Block = 256 threads = 8 waves. Each block handles 128 node rows; W is
// staged in LDS (64 KB). Each wave computes a 16x128 slab: 8 accumulator
// tiles of 16x16, K-loop of 32 steps of 4.
// -------------------------------------------------------------------------
__global__ __launch_bounds__(THREADS)
void gcn_gemm_wmma(const float* __restrict__ x,
                   const float* __restrict__ w,
                   float* __restrict__ h,
                   int n_nodes)
{
    __shared__ float wlds[D * D];   // 64 KB

    const int tid = threadIdx.x;

    // Stage W into LDS with float4 loads (4096 float4 / 256 threads = 16 each)
    {
        const float4* w4  = (const float4*)w;
        float4*       wl4 = (float4*)wlds;
#pragma unroll
        for (int i = 0; i < 16; ++i)
            wl4[tid + i * THREADS] = w4[tid + i * THREADS];
    }
    __syncthreads();

    const int wave = tid >> 5;       // 0..7
    const int lane = tid & 31;
    const int half = lane >> 4;      // 0: lanes 0-15, 1: lanes 16-31
    const int l16  = lane & 15;

    const int64_t row_base = (int64_t)blockIdx.x * ROWS_PER_BLOCK + wave * 16;

    // A-fragment row for this lane (clamped so EXEC stays all-ones for WMMA;
    // out-of-range results are masked at the store).
    int64_t arow = row_base + l16;
    if (arow >= n_nodes) arow = n_nodes - 1;
    const float* xrow = x + arow * D;

    v8f acc[8];
#pragma unroll
    for (int t = 0; t < 8; ++t)
#pragma unroll
        for (int i = 0; i < 8; ++i)
            acc[t][i] = 0.0f;

    // K loop: 128 in steps of 4. Lanes 0-15 carry K={k0,k0+1}, lanes 16-31
    // carry K={k0+2,k0+3} per the 32-bit 16x4 A layout.
    for (int kk = 0; kk < 32; ++kk) {
        const int k0 = kk * 4 + half * 2;

        v2f a;
        a.x = xrow[k0];
        a.y = xrow[k0 + 1];

#pragma unroll
        for (int t = 0; t < 8; ++t) {
            // B (4x16): VGPR0 = row k0 (lanes 0-15) / k0+2 (lanes 16-31),
            //           VGPR1 = row k0+1 / k0+3. k0 already folded in `half`.
            v2f b;
            b.x = wlds[k0 * D + t * 16 + l16];
            b.y = wlds[(k0 + 1) * D + t * 16 + l16];

            acc[t] = __builtin_amdgcn_wmma_f32_16x16x4_f32(
                /*neg_a=*/false, a, /*neg_b=*/false, b,
                /*c_mod=*/(short)0, acc[t],
                /*reuse_a=*/false, /*reuse_b=*/false);
        }
    }

    // Store D tiles: VGPR v -> row (v + half*8), col = l16 within the tile.
#pragma unroll
    for (int t = 0; t < 8; ++t) {
#pragma unroll
        for (int v = 0; v < 8; ++v) {
            const int64_t row = row_base + v + half * 8;
            if (row < n_nodes)
                h[row * D + t * 16 + l16] = acc[t][v];
        }
    }
}

// -------------------------------------------------------------------------
// Kernel 2: out[n][d] = bias[d]   (aggregation identity + bias fold)
// -------------------------------------------------------------------------
__global__ __launch_bounds__(THREADS)
void gcn_bias_init(const float* __restrict__ bias,
                   float* __restrict__ out,
                   int64_t total_f4)   // total floats / 4
{
    const int64_t i = (int64_t)blockIdx.x * blockDim.x + threadIdx.x;
    if (i >= total_f4) return;
    const int d = (int)((i * 4) & (D - 1));           // stays 16B-aligned
    *(float4*)(out + i * 4) = *(const float4*)(bias + d);
}

// -------------------------------------------------------------------------
// Kernel 3: scatter-add. One wave per edge; lane owns 4 consecutive features.
// Row gather is a contiguous 512B load spread over the wave; atomics land in
// L2 (51 MB output << 192 MB L2).
// -------------------------------------------------------------------------
__global__ __launch_bounds__(THREADS)
void gcn_scatter_add(const float* __restrict__ h,
                     const int64_t* __restrict__ src,
                     const int64_t* __restrict__ dst,
                     float* __restrict__ out,
                     int n_edges)
{
    const int64_t tid = (int64_t)blockIdx.x * blockDim.x + threadIdx.x;
    const int64_t e   = tid >> 5;
    if (e >= n_edges) return;
    const int lane = (int)(tid & 31);

    const int64_t s = src[e];   // uniform per wave -> scalar broadcast
    const int64_t d = dst[e];

    const float4 v = *(const float4*)(h + s * D + lane * 4);
    float* o = out + d * D + lane * 4;
    atomicAdd(o + 0, v.x);
    atomicAdd(o + 1, v.y);
    atomicAdd(o + 2, v.z);
    atomicAdd(o + 3, v.w);
}

// -------------------------------------------------------------------------
extern "C" void kernel_launch(void* const* d_in, const int* in_sizes, int n_in,
                              void* d_out, int out_size, void* d_ws, size_t ws_size,
                              hipStream_t stream)
{
    // setup_inputs() order: x [N,128] f32, edge_index [2,E] i64,
    //                       weight [128,128] f32, bias [128] f32
    const float*   x     = (const float*)d_in[0];
    const int64_t* eidx  = (const int64_t*)d_in[1];
    const float*   w     = (const float*)d_in[2];
    const float*   bias  = (const float*)d_in[3];
    float*         out   = (float*)d_out;

    const int n_nodes = in_sizes[0] / D;       // 100000
    const int n_edges = in_sizes[1] / 2;       // 1600000
    const int64_t* src = eidx;                 // edge_index[0]
    const int64_t* dst = eidx + n_edges;       // edge_index[1]

    float* h = (float*)d_ws;                   // [n_nodes, 128] fp32 scratch

    // 1) h = x @ W
    {
        const int grid = (n_nodes + ROWS_PER_BLOCK - 1) / ROWS_PER_BLOCK;
        gcn_gemm_wmma<<<grid, THREADS, 0, stream>>>(x, w, h, n_nodes);
    }

    // 2) out = bias (broadcast)
    {
        const int64_t total_f4 = (int64_t)n_nodes * D / 4;
        const int grid = (int)((total_f4 + THREADS - 1) / THREADS);
        gcn_bias_init<<<grid, THREADS, 0, stream>>>(bias, out, total_f4);
    }

    // 3) out[dst] += h[src]
    {
        const int64_t work = (int64_t)n_edges * 32;
        const int grid = (int)((work + THREADS - 1) / THREADS);
        gcn_scatter_add<<<grid, THREADS, 0, stream>>>(h, src, dst, out, n_edges);
    }
}